// GaussianOverlapLoss_19361712570495
// MI455X (gfx1250) — compile-verified
//
#include <hip/hip_runtime.h>
#include <math.h>

typedef __attribute__((ext_vector_type(2))) float v2f;
typedef __attribute__((ext_vector_type(8))) float v8f;

#define LAMB_C  0.0001f
#define ALPHA_C 0.01f
#define BETA_C  0.6065f
#define EPS_C   1e-07f
// -0.25 * log2(e): folds exp->exp2 conversion into the quad scale factor
#define NEG_QTR_LOG2E (-0.25f * 1.4426950408889634f)

// ---------------------------------------------------------------------------
// Kernel 1: per-Gaussian eigenvalue regularizer  sum log1p(sqrt(clip(eig)))
// ---------------------------------------------------------------------------
__global__ __launch_bounds__(256) void lamb_kernel(const float* __restrict__ sigma,
                                                   float* __restrict__ ws_lamb, int N) {
    __shared__ float red[256];
    int i = blockIdx.x * 256 + threadIdx.x;
    float val = 0.0f;
    if (i < N) {
        float a = sigma[4 * i + 0];
        float b = sigma[4 * i + 1];
        float c = sigma[4 * i + 3];
        float ht   = 0.5f * (a + c);
        float hd   = 0.5f * (a - c);
        float disc = __builtin_amdgcn_sqrtf(hd * hd + b * b);
        float e0 = fmaxf(ht - disc, EPS_C);
        float e1 = fmaxf(ht + disc, EPS_C);
        val = log1pf(__builtin_amdgcn_sqrtf(e0)) + log1pf(__builtin_amdgcn_sqrtf(e1));
    }
    red[threadIdx.x] = val;
    __syncthreads();
    for (int s = 128; s > 0; s >>= 1) {
        if (threadIdx.x < s) red[threadIdx.x] += red[threadIdx.x + s];
        __syncthreads();
    }
    if (threadIdx.x == 0) ws_lamb[blockIdx.x] = red[0];
}

// ---------------------------------------------------------------------------
// Kernel 2: pairwise overlap. One block per 16-row tile strip; 8 wave32s per
// block stride over the 256 column tiles. Bilinear forms T (rank 12),
// S = 4*detM (rank 5->8), P = det_s_i*det_s_j (rank 1->4) are computed with
// chained V_WMMA_F32_16X16X4_F32; VALU does only the transcendental epilogue
// using raw-rate v_rcp/v_sqrt/v_exp (1-ULP, fine for a loss scalar).
// ---------------------------------------------------------------------------
__global__ __launch_bounds__(256) void overlap_kernel(const float* __restrict__ mu,
                                                      const float* __restrict__ sigma,
                                                      const unsigned char* __restrict__ cmask,
                                                      float* __restrict__ ws_ov, int N) {
    __shared__ float red[256];

    const int tid   = threadIdx.x;
    const int lane  = tid & 31;
    const int wave  = tid >> 5;          // 0..7
    const int hi    = (lane >> 4) & 1;   // lane half selects K pair / row group
    const int n16   = lane & 15;
    const int tileI = blockIdx.x;
    const int tilesJ = N >> 4;

    // ---- row (i-side) features: point ri = 16*tileI + n16 ------------------
    const int ri = tileI * 16 + n16;
    const float xi = mu[2 * ri + 0], yi = mu[2 * ri + 1];
    const float ai = sigma[4 * ri + 0], bi = sigma[4 * ri + 1], ci = sigma[4 * ri + 3];
    const float xi2 = xi * xi, yi2 = yi * yi, xyi = xi * yi;
    const float pi_ = ci * xi2 - 2.0f * bi * xyi + ai * yi2;
    const float dsi = ai * ci - bi * bi;
    const float fi0 = 2.0f * (bi * yi - ci * xi);  // pairs with x_j
    const float fi1 = 2.0f * (bi * xi - ai * yi);  // pairs with y_j

    // A fragments (16x4 fp32: lanes 0-15 hold K={0,1}, lanes 16-31 hold K={2,3})
    // T features f0..f11:
    //  f0=2(by-cx) f1=2(bx-ay) f2=x f3=y | f4=c f5=a f6=b f7=x^2 |
    //  f8=y^2 f9=xy f10=p f11=1
    const v2f A0 = { hi ? xi  : fi0, hi ? yi  : fi1 };
    const v2f A1 = { hi ? bi  : ci,  hi ? xi2 : ai  };
    const v2f A2 = { hi ? pi_ : yi2, hi ? 1.0f: xyi };
    // S features f12..f19:  a, c, b, det_s, 1, 0, 0, 0
    const v2f A3 = { hi ? bi  : ai,  hi ? dsi : ci  };
    const v2f A4 = { hi ? 0.0f: 1.0f, 0.0f };
    // P features f20..f23:  det_s, 0, 0, 0
    const v2f A5 = { hi ? 0.0f: dsi, 0.0f };

    float partial = 0.0f;

    for (int tileJ = wave; tileJ < tilesJ; tileJ += 8) {
        // ---- column (j-side) features: point rj = 16*tileJ + n16 ----------
        const int rj = tileJ * 16 + n16;
        const float xj = mu[2 * rj + 0], yj = mu[2 * rj + 1];
        const float aj = sigma[4 * rj + 0], bj = sigma[4 * rj + 1], cj = sigma[4 * rj + 3];
        const float xj2 = xj * xj, yj2 = yj * yj, xyj = xj * yj;
        const float pj_ = cj * xj2 - 2.0f * bj * xyj + aj * yj2;
        const float dsj = aj * cj - bj * bj;
        const float gj2 = 2.0f * (bj * yj - cj * xj);  // pairs with x_i
        const float gj3 = 2.0f * (bj * xj - aj * yj);  // pairs with y_i

        // B fragments (4x16 fp32), g0..g11 for T:
        //  g0=x g1=y g2=2(by-cx) g3=2(bx-ay) | g4=x^2 g5=y^2 g6=-2xy g7=c |
        //  g8=a g9=-2b g10=1 g11=p
        const v2f B0 = { hi ? gj2        : xj,  hi ? gj3  : yj  };
        const v2f B1 = { hi ? -2.0f*xyj  : xj2, hi ? cj   : yj2 };
        const v2f B2 = { hi ? 1.0f       : aj,  hi ? pj_  : -2.0f*bj };
        // S: g12..g19 = c, a, -2b, 1, det_s, 0, 0, 0
        const v2f B3 = { hi ? -2.0f*bj   : cj,  hi ? 1.0f : aj  };
        const v2f B4 = { hi ? 0.0f       : dsj, 0.0f };
        // P: g20..g23 = det_s, 0, 0, 0
        const v2f B5 = { hi ? 0.0f       : dsj, 0.0f };

        v8f accT = {};
        accT = __builtin_amdgcn_wmma_f32_16x16x4_f32(false, A0, false, B0, (short)0, accT, false, false);
        accT = __builtin_amdgcn_wmma_f32_16x16x4_f32(false, A1, false, B1, (short)0, accT, false, false);
        accT = __builtin_amdgcn_wmma_f32_16x16x4_f32(false, A2, false, B2, (short)0, accT, false, false);
        v8f accS = {};
        accS = __builtin_amdgcn_wmma_f32_16x16x4_f32(false, A3, false, B3, (short)0, accS, false, false);
        accS = __builtin_amdgcn_wmma_f32_16x16x4_f32(false, A4, false, B4, (short)0, accS, false, false);
        v8f accP = {};
        accP = __builtin_amdgcn_wmma_f32_16x16x4_f32(false, A5, false, B5, (short)0, accP, false, false);

        // ---- epilogue: element (m,n) = (v + 8*hi, n16) of this 16x16 tile --
        const int jg = tileJ * 16 + n16;
        #pragma unroll
        for (int v = 0; v < 8; ++v) {
            const int m  = v + 8 * hi;
            const int ig = tileI * 16 + m;
            const float T = accT[v];
            const float S = accS[v];          // = 4*detM  (>0: PD average)
            const float P = accP[v];          // = det_s_i * det_s_j
            const float rS = __builtin_amdgcn_rcpf(S);                 // v_rcp_f32
            const float t0 = __builtin_amdgcn_exp2f(NEG_QTR_LOG2E * T * rS); // v_exp_f32
            const float t1 = __builtin_amdgcn_sqrtf(fmaxf(P, EPS_C));  // v_sqrt_f32
            float coef = t0 * __builtin_amdgcn_sqrtf(fmaxf(4.0f * t1 * rS, EPS_C));
            const unsigned char mb = cmask[(size_t)ig * (size_t)N + (size_t)jg];
            if (ig == jg || mb) coef = 0.0f;   // masked -> loss term exactly 0
            const float z = coef - BETA_C;
            partial += (z > 0.0f ? z : ALPHA_C * z) + BETA_C * ALPHA_C;
        }
    }

    red[tid] = partial;
    __syncthreads();
    for (int s = 128; s > 0; s >>= 1) {
        if (tid < s) red[tid] += red[tid + s];
        __syncthreads();
    }
    if (tid == 0) ws_ov[blockIdx.x] = red[0];
}

// ---------------------------------------------------------------------------
// Kernel 3: deterministic final combine -> scalar loss
// ---------------------------------------------------------------------------
__global__ void finalize_kernel(const float* __restrict__ ws, float* __restrict__ out,
                                int nLamb, int nOv, int N) {
    if (threadIdx.x == 0 && blockIdx.x == 0) {
        float sl = 0.0f, so = 0.0f;
        for (int k = 0; k < nLamb; ++k) sl += ws[k];
        for (int k = 0; k < nOv; ++k)  so += ws[nLamb + k];
        const float loss_lamb = LAMB_C * sl / (2.0f * (float)N);
        const float overlap   = so / (float)N;
        out[0] = loss_lamb + overlap;   // LOSS_WEIGHT = 1
    }
}

extern "C" void kernel_launch(void* const* d_in, const int* in_sizes, int n_in,
                              void* d_out, int out_size, void* d_ws, size_t ws_size,
                              hipStream_t stream) {
    const float* mu    = (const float*)d_in[0];
    const float* sigma = (const float*)d_in[1];
    const unsigned char* cmask = (const unsigned char*)d_in[2];
    float* out = (float*)d_out;
    float* ws  = (float*)d_ws;

    const int N = in_sizes[0] / 2;        // 4096
    const int nLamb = (N + 255) / 256;    // 16 partials
    const int nTiles = N / 16;            // 256 row-tile strips

    lamb_kernel<<<nLamb, 256, 0, stream>>>(sigma, ws, N);
    overlap_kernel<<<nTiles, 256, 0, stream>>>(mu, sigma, cmask, ws + nLamb, N);
    finalize_kernel<<<1, 1, 0, stream>>>(ws, out, nLamb, nTiles, N);
}